// GNNLayer_76459007803805
// MI455X (gfx1250) — compile-verified
//
#include <hip/hip_runtime.h>
#include <math.h>

typedef float v2f __attribute__((ext_vector_type(2)));
typedef float v8f __attribute__((ext_vector_type(8)));

#define Nn   6144
#define INF  256
#define OUTF 64
#define NH   4
#define LSTR 68   // padded LDS row stride (floats) -> conflict-free across rows

// D = A(16x4 f32) * B(4x16 f32) + C(16x16 f32), full f32 precision
static __device__ __forceinline__ v8f wmma4(v2f a, v2f b, v8f c) {
  return __builtin_amdgcn_wmma_f32_16x16x4_f32(
      /*neg_a=*/false, a, /*neg_b=*/false, b,
      /*c_mod=*/(short)0, c, /*reuse_a=*/false, /*reuse_b=*/false);
}

// gfx1250 has a hardware tanh transcendental; fall back to a branchless exp form.
static __device__ __forceinline__ float fast_tanh(float x) {
#if __has_builtin(__builtin_amdgcn_tanhf)
  return __builtin_amdgcn_tanhf(x);
#elif __has_builtin(__builtin_amdgcn_tanh_f32)
  return __builtin_amdgcn_tanh_f32(x);
#else
  float t = __expf(2.0f * fminf(fmaxf(x, -15.0f), 15.0f));
  return (t - 1.0f) / (t + 1.0f);
#endif
}

// Async-copy one 16-row x 64-float tile (row-major, contiguous rows of 64 f32)
// into LDS with padded row stride LSTR. One wave: 8 x b128 per tile
// (each instruction moves 2 rows: lanes 0-15 -> row 2m, lanes 16-31 -> row 2m+1).
static __device__ __forceinline__ void copy_tile_async(const float* lds_dst,
                                                       const float* gsrc,
                                                       int lane) {
  const int r0 = lane >> 4;          // 0..1
  const int c4 = (lane & 15) * 4;    // 16B chunk within a row
#pragma unroll
  for (int m = 0; m < 8; ++m) {
    const int row = m * 2 + r0;
    unsigned loff = (unsigned)(size_t)(lds_dst + row * LSTR + c4);
    unsigned long long g =
        (unsigned long long)(size_t)(gsrc + (size_t)row * OUTF + c4);
    asm volatile("global_load_async_to_lds_b128 %0, %1, off"
                 :: "v"(loff), "v"(g) : "memory");
  }
}

__global__ __launch_bounds__(32) void k_init(float* __restrict__ S) {
  if (threadIdx.x < NH) S[threadIdx.x] = 0.0f;
}

// f1 = tanh(X @ W1^T), f2 = tanh(X @ W2^T), f1a = active ? tanh(f1) : f1
__global__ __launch_bounds__(128) void k_fw(const float* __restrict__ X,
                                            const float* __restrict__ W1,
                                            const float* __restrict__ W2,
                                            const int*   __restrict__ act,
                                            float* __restrict__ F1,
                                            float* __restrict__ F2,
                                            float* __restrict__ F1A) {
  const int lane = threadIdx.x & 31;
  const int w    = threadIdx.x >> 5;
  const int h    = blockIdx.y;
  const int i0   = (blockIdx.x * 4 + w) * 16;
  const int cl   = lane & 15;
  const int kh   = (lane >> 4) * 2;

  const float* w1h = W1 + (size_t)h * OUTF * INF;
  const float* w2h = W2 + (size_t)h * OUTF * INF;

  v8f a1[4] = {}; v8f a2[4] = {};
  for (int k = 0; k < INF; k += 4) {
    v2f a = *(const v2f*)(X + (size_t)(i0 + cl) * INF + k + kh);
#pragma unroll
    for (int t = 0; t < 4; ++t) {
      const int n = t * 16 + cl;
      v2f b1 = *(const v2f*)(w1h + (size_t)n * INF + k + kh);
      v2f b2 = *(const v2f*)(w2h + (size_t)n * INF + k + kh);
      a1[t] = wmma4(a, b1, a1[t]);
      a2[t] = wmma4(a, b2, a2[t]);
    }
  }

  const int a_on = *act;
  float* f1h  = F1  + (size_t)h * Nn * OUTF;
  float* f2h  = F2  + (size_t)h * Nn * OUTF;
  float* f1ah = F1A + (size_t)h * Nn * OUTF;
#pragma unroll
  for (int t = 0; t < 4; ++t)
#pragma unroll
    for (int v = 0; v < 8; ++v) {
      const size_t r = (size_t)(i0 + v + (lane >> 4) * 8);
      const int    c = t * 16 + cl;
      float x1 = fast_tanh(a1[t][v]);
      f1h [r * OUTF + c] = x1;
      f1ah[r * OUTF + c] = a_on ? fast_tanh(x1) : x1;
      f2h [r * OUTF + c] = fast_tanh(a2[t][v]);
    }
}

// Fused pass: per 16-row block, stream column tiles (async double-buffered):
//   L = (f2_i @ f1_j^T)/8 ; M = exp(L)*adj ; T_i += M @ f1a_j ; S += sum(M)
__global__ __launch_bounds__(128) void k_attn(const float* __restrict__ adj,
                                              const float* __restrict__ F1,
                                              const float* __restrict__ F2,
                                              const float* __restrict__ F1A,
                                              float* __restrict__ T,
                                              float* __restrict__ S) {
  __shared__ float smF1 [2][16 * LSTR];   // block-shared f1 column tiles
  __shared__ float smF1A[2][16 * LSTR];   // block-shared f1a column tiles
  __shared__ float smStage[4][16 * 17];   // per-wave C->A relayout staging

  const int lane = threadIdx.x & 31;
  const int w    = threadIdx.x >> 5;
  const int h    = blockIdx.y;
  const int i0   = (blockIdx.x * 4 + w) * 16;
  const int cl   = lane & 15;
  const int kh   = (lane >> 4) * 2;

  const float* f1h  = F1  + (size_t)h * Nn * OUTF;
  const float* f2h  = F2  + (size_t)h * Nn * OUTF;
  const float* f1ah = F1A + (size_t)h * Nn * OUTF;

  // preload A = f2 row tile (16x64) in WMMA A layout
  v2f a2[16];
#pragma unroll
  for (int kk = 0; kk < 16; ++kk)
    a2[kk] = *(const v2f*)(f2h + (size_t)(i0 + cl) * OUTF + kk * 4 + kh);

  v8f   accT[4] = {};
  float lsum    = 0.0f;
  float* stg = &smStage[w][0];

  if (w == 0) {
    copy_tile_async(&smF1 [0][0], f1h,  lane);
    copy_tile_async(&smF1A[0][0], f1ah, lane);
  }

  for (int j0 = 0; j0 < Nn; j0 += 16) {
    const int cur = (j0 >> 4) & 1;
    if (w == 0) {
      if (j0 + 16 < Nn) {   // prefetch next tile, retire current one (in-order)
        copy_tile_async(&smF1 [cur ^ 1][0], f1h  + (size_t)(j0 + 16) * OUTF, lane);
        copy_tile_async(&smF1A[cur ^ 1][0], f1ah + (size_t)(j0 + 16) * OUTF, lane);
        asm volatile("s_wait_asynccnt 0x10" ::: "memory");
      } else {
        asm volatile("s_wait_asynccnt 0x0" ::: "memory");
      }
    }
    __syncthreads();                       // tile `cur` visible to all waves
    const float* tf1  = &smF1 [cur][0];
    const float* tf1a = &smF1A[cur][0];

    // logits tile from LDS; two accumulators break the WMMA RAW chain
    v8f c0 = {}, c1 = {};
#pragma unroll
    for (int kk = 0; kk < 16; kk += 2) {
      v2f b0 = *(const v2f*)(tf1 + cl * LSTR + (kk    ) * 4 + kh);
      v2f b1 = *(const v2f*)(tf1 + cl * LSTR + (kk + 1) * 4 + kh);
      c0 = wmma4(a2[kk],     b0, c0);
      c1 = wmma4(a2[kk + 1], b1, c1);
    }
    v8f c = c0 + c1;

    // masked exp + partial global sum (adj stays L2-resident)
    __builtin_prefetch(adj + (size_t)(i0 + cl) * Nn + j0 + 16, 0, 0);
#pragma unroll
    for (int v = 0; v < 8; ++v) {
      const size_t r = (size_t)(i0 + v + (lane >> 4) * 8);
      float e = __expf(c[v] * 0.125f) * adj[r * Nn + (size_t)(j0 + cl)];
      c[v] = e;
      lsum += e;
    }

    // stage C-layout tile -> LDS -> reload in A layout
#pragma unroll
    for (int v = 0; v < 8; ++v)
      stg[(v + (lane >> 4) * 8) * 17 + cl] = c[v];
    asm volatile("s_wait_dscnt 0x0" ::: "memory");

#pragma unroll
    for (int kk = 0; kk < 4; ++kk) {
      v2f ac;
      ac[0] = stg[cl * 17 + kk * 4 + kh];
      ac[1] = stg[cl * 17 + kk * 4 + kh + 1];
#pragma unroll
      for (int t = 0; t < 4; ++t) {
        v2f b;
        b[0] = tf1a[(kk * 4 + kh    ) * LSTR + t * 16 + cl];
        b[1] = tf1a[(kk * 4 + kh + 1) * LSTR + t * 16 + cl];
        accT[t] = wmma4(ac, b, accT[t]);
      }
    }
    __syncthreads();   // everyone done reading `cur` before wave0 overwrites it
  }

  float* Th = T + (size_t)h * Nn * OUTF;
#pragma unroll
  for (int t = 0; t < 4; ++t)
#pragma unroll
    for (int v = 0; v < 8; ++v)
      Th[(size_t)(i0 + v + (lane >> 4) * 8) * OUTF + t * 16 + cl] = accT[t][v];

#pragma unroll
  for (int off = 16; off; off >>= 1) lsum += __shfl_xor(lsum, off, 32);
  if (lane == 0) atomicAdd(&S[h], lsum);
}

// out = sum_h (adj @ T_h) * (0.25 / S_h), T chunks async double-buffered in LDS
__global__ __launch_bounds__(128) void k_out(const float* __restrict__ adj,
                                             const float* __restrict__ T,
                                             const float* __restrict__ S,
                                             float* __restrict__ out) {
  __shared__ float smT[2][16 * LSTR];
  const int lane = threadIdx.x & 31;
  const int w    = threadIdx.x >> 5;
  const int i0   = (blockIdx.x * 4 + w) * 16;
  const int cl   = lane & 15;
  const int kh   = (lane >> 4) * 2;

  v8f tot[4] = {};
  for (int h = 0; h < NH; ++h) {
    const float* Th  = T + (size_t)h * Nn * OUTF;
    const float  inv = 0.25f / S[h];
    v8f acc[4] = {};

    if (w == 0) copy_tile_async(&smT[0][0], Th, lane);

    for (int k0 = 0; k0 < Nn; k0 += 16) {
      const int cur = (k0 >> 4) & 1;
      if (w == 0) {
        if (k0 + 16 < Nn) {
          copy_tile_async(&smT[cur ^ 1][0], Th + (size_t)(k0 + 16) * OUTF, lane);
          asm volatile("s_wait_asynccnt 0x8" ::: "memory");
        } else {
          asm volatile("s_wait_asynccnt 0x0" ::: "memory");
        }
      }
      __syncthreads();
      const float* tt = &smT[cur][0];
#pragma unroll
      for (int ks = 0; ks < 4; ++ks) {
        v2f a = *(const v2f*)(adj + (size_t)(i0 + cl) * Nn + k0 + ks * 4 + kh);
#pragma unroll
        for (int t = 0; t < 4; ++t) {
          v2f b;
          b[0] = tt[(ks * 4 + kh    ) * LSTR + t * 16 + cl];
          b[1] = tt[(ks * 4 + kh + 1) * LSTR + t * 16 + cl];
          acc[t] = wmma4(a, b, acc[t]);
        }
      }
      __syncthreads();
    }
#pragma unroll
    for (int t = 0; t < 4; ++t) tot[t] += acc[t] * inv;
  }

#pragma unroll
  for (int t = 0; t < 4; ++t)
#pragma unroll
    for (int v = 0; v < 8; ++v)
      out[(size_t)(i0 + v + (lane >> 4) * 8) * OUTF + t * 16 + cl] = tot[t][v];
}

extern "C" void kernel_launch(void* const* d_in, const int* in_sizes, int n_in,
                              void* d_out, int out_size, void* d_ws, size_t ws_size,
                              hipStream_t stream) {
  const float* X   = (const float*)d_in[0];   // [6144, 256]
  const float* adj = (const float*)d_in[1];   // [6144, 6144]
  const float* W1  = (const float*)d_in[2];   // [4, 64, 256]
  const float* W2  = (const float*)d_in[3];   // [4, 64, 256]
  const int*   act = (const int*)  d_in[4];   // scalar flag
  float* out = (float*)d_out;                 // [6144, 64]

  float* ws  = (float*)d_ws;
  float* Sn  = ws;                            // 4 floats (padded to 64)
  float* F1  = ws + 64;                       // [4][6144][64]
  float* F2  = F1 + (size_t)NH * Nn * OUTF;
  float* F1A = F2 + (size_t)NH * Nn * OUTF;
  float* T   = F1A + (size_t)NH * Nn * OUTF;  // total ~25 MB of ws

  const int rb = Nn / 64;                     // 96 row-block groups (4 waves each)
  k_init<<<1, 32, 0, stream>>>(Sn);
  k_fw  <<<dim3(rb, NH), 128, 0, stream>>>(X, W1, W2, act, F1, F2, F1A);
  k_attn<<<dim3(rb, NH), 128, 0, stream>>>(adj, F1, F2, F1A, T, Sn);
  k_out <<<dim3(rb),     128, 0, stream>>>(adj, T, Sn, out);
}